// QualityAwarePrompt_85409719649041
// MI455X (gfx1250) — compile-verified
//
#include <hip/hip_runtime.h>
#include <hip/hip_bf16.h>

#define H_DIM   256
#define D_DIM   512
#define POOL    20
#define TOPK    5
#define LMIN    5
#define LMAX    64
#define PN      (LMAX * D_DIM)   // 32768 flattened (L, D)
#define NTILES  (PN / 16)        // 2048 column tiles
#define NTGRP   (NTILES / 8)     // 256 column-tile groups (8 waves/block)

typedef __attribute__((ext_vector_type(2))) float v2f;
typedef __attribute__((ext_vector_type(8))) float v8f;

__device__ __forceinline__ float block_reduce_sum(float* red, int tid, float v) {
    red[tid] = v;
    __syncthreads();
    for (int s = 128; s > 0; s >>= 1) {
        if (tid < s) red[tid] += red[tid + s];
        __syncthreads();
    }
    float r = red[0];
    __syncthreads();
    return r;
}

// Block 0: scale = 1 + 0.5 * mean(quality).  Blocks 1..POOL: inverse key norms.
__global__ __launch_bounds__(256)
void qap_prep_kernel(const float* __restrict__ qual,
                     const float* __restrict__ keys,
                     float* __restrict__ scale_out,
                     float* __restrict__ inv_knorm,
                     int B) {
    __shared__ float red[256];
    int tid = threadIdx.x;
    if (blockIdx.x == 0) {
        float s = 0.0f;
        for (int i = tid; i < B; i += 256) s += qual[i];
        float tot = block_reduce_sum(red, tid, s);
        if (tid == 0) scale_out[0] = 1.0f + 0.5f * (tot / (float)B);
    } else {
        int p = blockIdx.x - 1;
        float v0 = keys[p * D_DIM + tid];
        float v1 = keys[p * D_DIM + tid + 256];
        float ss = block_reduce_sum(red, tid, v0 * v0 + v1 * v1);
        if (tid == 0) inv_knorm[p] = 1.0f / fmaxf(sqrtf(ss), 1e-8f);
    }
}

// One block per sample: MLP projection -> LN -> ReLU -> GEMV -> cosine sims
// -> scaled softmax -> top-5 as dense sparse weight row W20[b, 0..19] + lens[b].
__global__ __launch_bounds__(256)
void qap_proj_kernel(const float* __restrict__ qual,
                     const float* __restrict__ w1, const float* __restrict__ b1,
                     const float* __restrict__ ln_g, const float* __restrict__ ln_b,
                     const float* __restrict__ w2, const float* __restrict__ b2,
                     const float* __restrict__ keys,
                     const float* __restrict__ scale_p,
                     const float* __restrict__ iknorm,
                     float* __restrict__ W20, int* __restrict__ lens, int B) {
    __shared__ float red[256];
    __shared__ float hbuf[H_DIM];
    __shared__ float qbuf[D_DIM];
    __shared__ float sims[POOL];
    int tid = threadIdx.x;
    int bb  = blockIdx.x;
    if (bb >= B) return;
    float qv = qual[bb];

    // h = q*w1 + b1 ; LayerNorm ; ReLU   (H_DIM == blockDim.x == 256)
    float hv  = qv * w1[tid] + b1[tid];
    float mu  = block_reduce_sum(red, tid, hv) * (1.0f / (float)H_DIM);
    float dv  = hv - mu;
    float var = block_reduce_sum(red, tid, dv * dv) * (1.0f / (float)H_DIM);
    float hn  = dv * rsqrtf(var + 1e-5f) * ln_g[tid] + ln_b[tid];
    hn = fmaxf(hn, 0.0f);
    hbuf[tid] = hn;
    __syncthreads();

    // query = h @ w2 + b2  (each thread owns 2 of 512 dims; coalesced w2 reads)
    float q0 = b2[tid], q1 = b2[tid + 256];
    for (int i = 0; i < H_DIM; ++i) {
        float hi = hbuf[i];
        q0 = fmaf(hi, w2[i * D_DIM + tid],        q0);
        q1 = fmaf(hi, w2[i * D_DIM + tid + 256],  q1);
    }
    qbuf[tid] = q0;
    qbuf[tid + 256] = q1;
    float nsq  = block_reduce_sum(red, tid, q0 * q0 + q1 * q1); // syncs -> qbuf visible
    float qinv = 1.0f / fmaxf(sqrtf(nsq), 1e-8f);

    // cosine similarities vs the 20 pool keys
    for (int p = 0; p < POOL; ++p) {
        float part = qbuf[tid] * keys[p * D_DIM + tid]
                   + qbuf[tid + 256] * keys[p * D_DIM + tid + 256];
        float s = block_reduce_sum(red, tid, part);
        if (tid == 0) sims[p] = s * qinv * iknorm[p];
    }
    __syncthreads();

    if (tid == 0) {
        float sc = scale_p[0];
        float z[POOL];
        float mx = -1e30f;
        for (int p = 0; p < POOL; ++p) { z[p] = sims[p] * sc; mx = fmaxf(mx, z[p]); }
        float se = 0.0f;
        for (int p = 0; p < POOL; ++p) { z[p] = expf(z[p] - mx); se += z[p]; }
        float inv = 1.0f / se;
        float wgt[POOL];
        for (int p = 0; p < POOL; ++p) wgt[p] = z[p] * inv;
        // top-5 (first occurrence on ties, matching lax.top_k), others zero
        bool  used[POOL];
        float outw[POOL];
        for (int p = 0; p < POOL; ++p) { used[p] = false; outw[p] = 0.0f; }
        for (int k = 0; k < TOPK; ++k) {
            int best = -1; float bv = -1.0f;
            for (int p = 0; p < POOL; ++p)
                if (!used[p] && wgt[p] > bv) { bv = wgt[p]; best = p; }
            used[best] = true;
            outw[best] = wgt[best];
        }
        for (int p = 0; p < POOL; ++p) W20[bb * POOL + p] = outw[p];
        // dynamic length
        float lf = (float)LMIN + (float)(LMAX - LMIN) * (1.0f - qv * 0.2f);
        int li = (int)lf; // trunc (lf > 0)
        li = li < LMIN ? LMIN : (li > LMAX ? LMAX : li);
        lens[bb] = li;
    }
}

// out[B, PN] = (W[B,20] @ PE[20,PN]) masked by per-row length.
// One wave32 per 16x16 tile; K=20 -> five V_WMMA_F32_16X16X4_F32 steps.
// blockIdx encodes (row tile, column-tile group) so m0 is block-uniform:
// the row-bounds test is a single scalar branch; the common full-tile path
// has zero exec-mask manipulation, int4 lens loads, and clause-able NT stores.
__global__ __launch_bounds__(256)
void qap_wmma_out_kernel(const float* __restrict__ W,
                         const int* __restrict__ lens,
                         const float* __restrict__ PE,
                         float* __restrict__ out,
                         int B) {
    int lane = threadIdx.x & 31;
    int wave = threadIdx.x >> 5;
    int mt   = blockIdx.x / NTGRP;            // block-uniform row tile
    int nt   = (blockIdx.x % NTGRP) * 8 + wave;
    int m0 = mt << 4, n0 = nt << 4;

    int r  = lane & 15;                       // A: row M ; B: col N ; D: col N
    int kb = (lane < 16) ? 0 : 2;             // A/B K sub-pair per lane half
    int arow = m0 + r;
    int crow = arow < B ? arow : (B - 1);     // clamp: only matters in tail tile

    const float* wrow = W  + (long)crow * POOL + kb;
    const float* bcol = PE + (long)kb * PN + n0 + r;

    v2f af[5], bf[5];
#pragma unroll
    for (int s = 0; s < 5; ++s) {
        af[s].x = wrow[s * 4];
        af[s].y = wrow[s * 4 + 1];
        bf[s].x = bcol[(long)(s * 4) * PN];
        bf[s].y = bcol[(long)(s * 4 + 1) * PN];
    }

    v8f acc = {};
#pragma unroll
    for (int s = 0; s < 5; ++s)
        acc = __builtin_amdgcn_wmma_f32_16x16x4_f32(
            /*neg_a=*/false, af[s], /*neg_b=*/false, bf[s],
            /*c_mod=*/(short)0, acc, /*reuse_a=*/false, /*reuse_b=*/false);

    int l = n0 >> 9;                          // column tile lies in one L slot
    int mbase = (lane < 16) ? 0 : 8;
    float* orow = out + (long)(m0 + mbase) * PN + n0 + r;

    if (m0 + 16 <= B) {                       // uniform predicate -> scalar branch
        const int4* lp = (const int4*)(lens + m0 + mbase);  // 16B-aligned
        int4 la = lp[0], lb = lp[1];
        int lv[8] = { la.x, la.y, la.z, la.w, lb.x, lb.y, lb.z, lb.w };
#pragma unroll
        for (int v = 0; v < 8; ++v) {
            float val = (l < lv[v]) ? acc[v] : 0.0f;
            __builtin_nontemporal_store(val, orow + (long)v * PN);
        }
    } else {                                  // rare tail tile (B % 16 != 0)
#pragma unroll
        for (int v = 0; v < 8; ++v) {
            int row = m0 + mbase + v;
            if (row < B) {
                float val = (l < lens[row]) ? acc[v] : 0.0f;
                __builtin_nontemporal_store(val, orow + (long)v * PN);
            }
        }
    }
}

extern "C" void kernel_launch(void* const* d_in, const int* in_sizes, int n_in,
                              void* d_out, int out_size, void* d_ws, size_t ws_size,
                              hipStream_t stream) {
    // Input order: x_embed(0, unused), quality(1), keys(2), prompt_embeddings(3),
    //              w1(4), b1(5), ln_g(6), ln_b(7), w2(8), b2(9)
    const float* qual = (const float*)d_in[1];
    const float* keys = (const float*)d_in[2];
    const float* pe   = (const float*)d_in[3];
    const float* w1   = (const float*)d_in[4];
    const float* b1   = (const float*)d_in[5];
    const float* lng  = (const float*)d_in[6];
    const float* lnb  = (const float*)d_in[7];
    const float* w2   = (const float*)d_in[8];
    const float* b2   = (const float*)d_in[9];
    float* out = (float*)d_out;
    int B = in_sizes[1];  // quality_score flat count == batch

    float* ws_f    = (float*)d_ws;
    float* scale   = ws_f;              // 1 float
    float* iknorm  = ws_f + 1;          // POOL floats
    float* W20     = ws_f + 32;         // B*POOL floats (offset keeps lens 16B-aligned)
    int*   lens    = (int*)(ws_f + 32 + (size_t)B * POOL);

    qap_prep_kernel<<<1 + POOL, 256, 0, stream>>>(qual, keys, scale, iknorm, B);
    qap_proj_kernel<<<B, 256, 0, stream>>>(qual, w1, b1, lng, lnb, w2, b2,
                                           keys, scale, iknorm, W20, lens, B);
    int MT = (B + 15) / 16;
    qap_wmma_out_kernel<<<MT * NTGRP, 256, 0, stream>>>(W20, lens, pe, out, B);
}